// Discriminator_19533511262893
// MI455X (gfx1250) — compile-verified
//
#include <hip/hip_runtime.h>
#include <math.h>

typedef __attribute__((ext_vector_type(2))) float v2f;
typedef __attribute__((ext_vector_type(8))) float v8f;

#define T_LEN  512
#define BATCH  64
#define HID    100
#define HID2   200
#define NPAD   112          // 7 tiles of 16 covering HID=100 (B/W padding only)
#define NLAYER 10
#define TBROWS (T_LEN * BATCH)
#define MROWS  16           // batch rows per scan block

// ---- CDNA5 async global->LDS path (ASYNCcnt) with sync fallback -----------
#if __has_builtin(__builtin_amdgcn_global_load_async_to_lds_b128) && \
    __has_builtin(__builtin_amdgcn_s_wait_asynccnt)
typedef int async_v4i __attribute__((vector_size(16)));
#define ASYNC_CP16(g, l)                                                    \
    __builtin_amdgcn_global_load_async_to_lds_b128(                         \
        (__attribute__((address_space(1))) async_v4i*)(g),                  \
        (__attribute__((address_space(3))) async_v4i*)(l), 0, 0)
#define WAIT_ASYNC(n) __builtin_amdgcn_s_wait_asynccnt(n)
#else
#define ASYNC_CP16(g, l) (*(float4*)(l) = *(const float4*)(g))
#define WAIT_ASYNC(n) ((void)0)
#endif

// ---- hardware tanh (TRANS op on CDNA5) with libm fallback ------------------
#if __has_builtin(__builtin_amdgcn_tanhf)
#define TANH(x) __builtin_amdgcn_tanhf(x)
#else
#define TANH(x) tanhf(x)
#endif

// ---------------------------------------------------------------------------
// Input projection: z[dir][t*B+b][h] = sum_i in[t*B+b][i] * Wih[dir][h][i]
//                                      + b_ih[dir][h] + b_hh[dir][h]
// One block = one 64-row x 112-col tile for one direction. fp32 WMMA 16x16x4.
// ---------------------------------------------------------------------------
template <int IN>
__global__ __launch_bounds__(256) void rnn_proj_kernel(
    const float* __restrict__ in,    // [TB][IN]
    const float* __restrict__ Wih,   // [2][HID][IN]
    const float* __restrict__ bi,    // [2][HID]
    const float* __restrict__ bh,    // [2][HID]
    float* __restrict__ z)           // [2][TB][HID]
{
    __shared__ __align__(16) float sX[64][IN];  // input tile (row-major)
    __shared__ float sW[IN][NPAD];              // W^T, zero-padded cols

    const int dir = blockIdx.y;
    const int r0  = blockIdx.x * 64; // row base into flattened T*B
    const int tid = threadIdx.x;

    // Async-stage the contiguous 64xIN input tile into LDS
    {
        const char* src = (const char*)(in + (size_t)r0 * IN);
        char* dst = (char*)&sX[0][0];
        for (int c = tid; c < 16 * IN; c += 256)      // 16B chunks
            ASYNC_CP16(src + (size_t)c * 16, dst + (size_t)c * 16);
    }
    // Stage W^T (sW[i][h] = Wih[dir][h][i]); pad h>=100 with zeros
    const float* Wd = Wih + (size_t)dir * HID * IN;
    for (int idx = tid; idx < IN * NPAD; idx += 256) {
        int i = idx / NPAD, h = idx % NPAD;
        sW[i][h] = (h < HID) ? Wd[(size_t)h * IN + i] : 0.0f;
    }
    WAIT_ASYNC(0);
    __syncthreads();

    const int wave  = tid >> 5;
    const int lane  = tid & 31;
    const int lhalf = lane >> 4;  // 0/1
    const int l16   = lane & 15;

    float* zd = z + ((size_t)dir * TBROWS + r0) * HID;

    // 4 M-tiles x 7 N-tiles = 28 tiles over 8 waves
    for (int tile = wave; tile < 28; tile += 8) {
        const int m0 = (tile / 7) * 16;
        const int n0 = (tile % 7) * 16;
        const int arow = m0 + l16;
        const int bcol = n0 + l16;

        v8f acc0 = {}, acc1 = {};
        constexpr int KS = IN / 4;       // 25 or 50
        constexpr int KH = KS / 2;
#pragma unroll
        for (int kk = 0; kk < KH; ++kk) {
            const int k = kk * 4 + 2 * lhalf;
            v2f a, b;
            a.x = sX[arow][k];     a.y = sX[arow][k + 1];
            b.x = sW[k][bcol];     b.y = sW[k + 1][bcol];
            acc0 = __builtin_amdgcn_wmma_f32_16x16x4_f32(false, a, false, b,
                                                         (short)0, acc0, false, false);
        }
#pragma unroll
        for (int kk = KH; kk < KS; ++kk) {
            const int k = kk * 4 + 2 * lhalf;
            v2f a, b;
            a.x = sX[arow][k];     a.y = sX[arow][k + 1];
            b.x = sW[k][bcol];     b.y = sW[k + 1][bcol];
            acc1 = __builtin_amdgcn_wmma_f32_16x16x4_f32(false, a, false, b,
                                                         (short)0, acc1, false, false);
        }
        v8f acc = acc0 + acc1;

        const int col = n0 + l16;
        if (col < HID) {
            const float bias = bi[dir * HID + col] + bh[dir * HID + col];
#pragma unroll
            for (int v = 0; v < 8; ++v) {
                const int row = m0 + v + 8 * lhalf;
                zd[(size_t)row * HID + col] = acc[v] + bias;
            }
        }
    }
}

// ---------------------------------------------------------------------------
// Sequential recurrence: h_t = tanh(z_t + h_{t-1} @ Whh^T), 512 steps.
// Grid = (4 batch-slices, 2 directions); each block owns a 16-row batch slice
// (the recurrence is independent per batch row). One 16x16 tile per wave,
// K=100 split into 4 WMMA accumulator chains to minimize serial depth.
// z is double-buffered in LDS via async global->LDS copies.
// ---------------------------------------------------------------------------
__global__ __launch_bounds__(224) void rnn_scan_kernel(
    const float* __restrict__ z,    // [2][T][B][HID]
    const float* __restrict__ Whh,  // [2][HID][HID]
    float* __restrict__ out)        // [T][B][2H] (this dir writes cols dir*H..)
{
    __shared__ float sW[HID][NPAD];                    // Whh^T, padded cols
    __shared__ __align__(16) float sH[2][MROWS][HID];  // ping-pong hidden
    __shared__ __align__(16) float sZ[2][MROWS * HID]; // double-buffered z_t

    const int mb0 = blockIdx.x * MROWS;  // batch-row base
    const int dir = blockIdx.y;
    const int tid = threadIdx.x;         // 224 threads = 7 waves

    const float* W = Whh + (size_t)dir * HID * HID;
    for (int idx = tid; idx < HID * NPAD; idx += 224) {
        int k = idx / NPAD, n = idx % NPAD;
        sW[k][n] = (n < HID) ? W[(size_t)n * HID + k] : 0.0f;
    }
    for (int idx = tid; idx < 2 * MROWS * HID; idx += 224)
        (&sH[0][0][0])[idx] = 0.0f;

    // z rows for this block are contiguous: 16*100 floats per timestep.
    const float* zblk = z + ((size_t)dir * T_LEN * BATCH + mb0) * HID;

    // Prologue: stage z for the first timestep into sZ[0] (400 x 16B chunks)
    {
        const int t0 = dir ? (T_LEN - 1) : 0;
        const char* src = (const char*)(zblk + (size_t)t0 * BATCH * HID);
        char* dst = (char*)&sZ[0][0];
        if (tid < 200) {
            ASYNC_CP16(src + (size_t)tid * 32, dst + (size_t)tid * 32);
            ASYNC_CP16(src + (size_t)tid * 32 + 16, dst + (size_t)tid * 32 + 16);
        }
    }
    WAIT_ASYNC(0);
    __syncthreads();

    const int wave  = tid >> 5;   // 0..6 == N-tile index
    const int lane  = tid & 31;
    const int lhalf = lane >> 4;
    const int l16   = lane & 15;
    const int n0    = wave * 16;
    const int arow  = l16;        // local batch row for A fragment
    const int bcol  = n0 + l16;

    int buf = 0;
    for (int step = 0; step < T_LEN; ++step) {
        const int t = dir ? (T_LEN - 1 - step) : step;

        // Kick off async staging of next timestep's z into the other buffer
        if (step + 1 < T_LEN) {
            const int tn = dir ? (t - 1) : (t + 1);
            const char* src = (const char*)(zblk + (size_t)tn * BATCH * HID);
            char* dst = (char*)&sZ[buf ^ 1][0];
            if (tid < 200) {
                ASYNC_CP16(src + (size_t)tid * 32, dst + (size_t)tid * 32);
                ASYNC_CP16(src + (size_t)tid * 32 + 16, dst + (size_t)tid * 32 + 16);
            }
        }

        // 16x16 tile: K=100 as 4 independent accumulator chains (depth <= 7)
        const float(*hc)[HID] = sH[buf];
        v8f a0 = {}, a1 = {}, a2 = {}, a3 = {};
#pragma unroll
        for (int kk = 0; kk < 7; ++kk) {
            const int k = kk * 4 + 2 * lhalf;
            v2f a, b;
            a.x = hc[arow][k];   a.y = hc[arow][k + 1];
            b.x = sW[k][bcol];   b.y = sW[k + 1][bcol];
            a0 = __builtin_amdgcn_wmma_f32_16x16x4_f32(false, a, false, b,
                                                       (short)0, a0, false, false);
        }
#pragma unroll
        for (int kk = 7; kk < 13; ++kk) {
            const int k = kk * 4 + 2 * lhalf;
            v2f a, b;
            a.x = hc[arow][k];   a.y = hc[arow][k + 1];
            b.x = sW[k][bcol];   b.y = sW[k + 1][bcol];
            a1 = __builtin_amdgcn_wmma_f32_16x16x4_f32(false, a, false, b,
                                                       (short)0, a1, false, false);
        }
#pragma unroll
        for (int kk = 13; kk < 19; ++kk) {
            const int k = kk * 4 + 2 * lhalf;
            v2f a, b;
            a.x = hc[arow][k];   a.y = hc[arow][k + 1];
            b.x = sW[k][bcol];   b.y = sW[k + 1][bcol];
            a2 = __builtin_amdgcn_wmma_f32_16x16x4_f32(false, a, false, b,
                                                       (short)0, a2, false, false);
        }
#pragma unroll
        for (int kk = 19; kk < 25; ++kk) {
            const int k = kk * 4 + 2 * lhalf;
            v2f a, b;
            a.x = hc[arow][k];   a.y = hc[arow][k + 1];
            b.x = sW[k][bcol];   b.y = sW[k + 1][bcol];
            a3 = __builtin_amdgcn_wmma_f32_16x16x4_f32(false, a, false, b,
                                                       (short)0, a3, false, false);
        }
        v8f acc = (a0 + a1) + (a2 + a3);

        // Ensure this timestep's z staging (issued last step) is complete in
        // *all* lanes' LDS view before the epilogue reads it.
        if (step + 1 < T_LEN) { WAIT_ASYNC(2); } else { WAIT_ASYNC(0); }
        __syncthreads();

        const int col = n0 + l16;
        float* od = out + ((size_t)t * BATCH + mb0) * HID2 + dir * HID;
        if (col < HID) {
#pragma unroll
            for (int v = 0; v < 8; ++v) {
                const int row = v + 8 * lhalf;  // local row
                const float val = TANH(acc[v] + sZ[buf][row * HID + col]);
                sH[buf ^ 1][row][col] = val;
                od[(size_t)row * HID2 + col] = val;
            }
        }
        __syncthreads();
        buf ^= 1;
    }
}

// ---------------------------------------------------------------------------
// Output head: y = sigmoid(h @ W_out^T + b_out), one thread per (t,b)
// ---------------------------------------------------------------------------
__global__ __launch_bounds__(256) void out_proj_kernel(
    const float* __restrict__ h,      // [TB][2H]
    const float* __restrict__ W_out,  // [2H]
    const float* __restrict__ b_out,  // [1]
    float* __restrict__ y)            // [TB]
{
    const int idx = blockIdx.x * blockDim.x + threadIdx.x;
    if (idx >= TBROWS) return;
    const float* hp = h + (size_t)idx * HID2;
    float s = b_out[0];
#pragma unroll 4
    for (int i = 0; i < HID2; ++i) s += hp[i] * W_out[i];
    y[idx] = 1.0f / (1.0f + __expf(-s));
}

// ---------------------------------------------------------------------------
extern "C" void kernel_launch(void* const* d_in, const int* in_sizes, int n_in,
                              void* d_out, int out_size, void* d_ws, size_t ws_size,
                              hipStream_t stream) {
    const float* x     = (const float*)d_in[0];  // [T,B,H]
    const float* W_ih0 = (const float*)d_in[1];  // [2,H,H]
    const float* W_ih  = (const float*)d_in[2];  // [L-1,2,H,2H]
    const float* W_hh  = (const float*)d_in[3];  // [L,2,H,H]
    const float* b_ih  = (const float*)d_in[4];  // [L,2,H]
    const float* b_hh  = (const float*)d_in[5];  // [L,2,H]
    const float* W_out = (const float*)d_in[6];  // [1,2H]
    const float* b_out = (const float*)d_in[7];  // [1]
    float* y = (float*)d_out;

    // Workspace: two [TB][2H] ping-pong activation buffers + z [2][TB][H]
    float* bufA = (float*)d_ws;
    float* bufB = bufA + (size_t)TBROWS * HID2;
    float* zbuf = bufB + (size_t)TBROWS * HID2;

    float* cur = bufA;
    float* nxt = bufB;
    for (int l = 0; l < NLAYER; ++l) {
        const float* bi = b_ih + (size_t)l * 2 * HID;
        const float* bh = b_hh + (size_t)l * 2 * HID;
        if (l == 0) {
            rnn_proj_kernel<HID><<<dim3(TBROWS / 64, 2), 256, 0, stream>>>(
                x, W_ih0, bi, bh, zbuf);
        } else {
            rnn_proj_kernel<HID2><<<dim3(TBROWS / 64, 2), 256, 0, stream>>>(
                cur, W_ih + (size_t)(l - 1) * 2 * HID * HID2, bi, bh, zbuf);
        }
        rnn_scan_kernel<<<dim3(BATCH / MROWS, 2), 224, 0, stream>>>(
            zbuf, W_hh + (size_t)l * 2 * HID * HID, nxt);
        float* tmp = cur; cur = nxt; nxt = tmp;
    }
    out_proj_kernel<<<(TBROWS + 255) / 256, 256, 0, stream>>>(cur, W_out, b_out, y);
}